// NodeAggrMLP_12541304504451
// MI455X (gfx1250) — compile-verified
//
#include <hip/hip_runtime.h>

typedef float v2f __attribute__((ext_vector_type(2)));
typedef float v4f __attribute__((ext_vector_type(4)));
typedef float v8f __attribute__((ext_vector_type(8)));

#define DZ 64   // z feature dim (== x feature dim)
#define H  8    // output head dim

// ---------------------------------------------------------------------------
// Kernel 1: tiny projections zh = z@Wz + b, xh = x@Wx via V_WMMA_F32_16X16X4_F32
// One wave32 per 16-row tile. K=64 -> 16 chained accumulating WMMAs.
// N dimension (H=8) padded to 16 with zero columns in the B fragment; the
// padding is done with a hoisted 0/1 mask + clamped column index so the loop
// body is branch-free (no exec save/restore between WMMAs).
// ---------------------------------------------------------------------------
__global__ void __launch_bounds__(32) proj_wmma(
    const float* __restrict__ z, const float* __restrict__ x,
    const float* __restrict__ W, const float* __restrict__ bias,
    float* __restrict__ zh, float* __restrict__ xh, int btiles)
{
    const int lane = threadIdx.x;          // 0..31, full wave, EXEC all 1s
    const int tile = blockIdx.x;
    const bool isX = tile >= btiles;
    const float* __restrict__ src = isX ? x : z;
    const float* __restrict__ Wp  = W + (isX ? DZ * H : 0);   // Wz rows 0..63, Wx rows 64..127
    const int row0 = (isX ? tile - btiles : tile) * 16;

    const int m  = lane & 15;              // A row / B,C,D column
    const int hi = lane >> 4;              // half-wave selector
    const int col = m & 7;                 // clamped W column (always in-bounds)
    const float msk = (m < H) ? 1.0f : 0.0f;   // zero-pad columns 8..15

    v8f acc = {0.f, 0.f, 0.f, 0.f, 0.f, 0.f, 0.f, 0.f};

    // A fragment base: A[m][k0 + 2*hi], A[m][k0 + 2*hi + 1]
    const float* __restrict__ arow = src + (size_t)(row0 + m) * DZ + 2 * hi;
    const float* __restrict__ wrow = Wp + (size_t)(2 * hi) * H + col;

    #pragma unroll
    for (int k0 = 0; k0 < DZ; k0 += 4) {
        v2f a, bf;
        a.x = arow[k0];
        a.y = arow[k0 + 1];
        // B fragment: B[k0 + 2*hi][n], B[k0 + 2*hi + 1][n], n = lane&15 (masked)
        bf.x = wrow[(size_t)k0 * H]     * msk;
        bf.y = wrow[(size_t)k0 * H + H] * msk;
        // 8 args: (neg_a, A, neg_b, B, c_mod, C, reuse_a, reuse_b)
        acc = __builtin_amdgcn_wmma_f32_16x16x4_f32(
            false, a, false, bf, (short)0, acc, false, false);
    }

    // D layout: acc[v] = out[row0 + v + 8*hi][n = lane&15]; only n < H valid.
    float* __restrict__ dst = isX ? xh : zh;
    if (m < H) {
        const float badd = isX ? 0.f : bias[m];   // fold bias into zh once
        #pragma unroll
        for (int v = 0; v < 8; v++) {
            dst[(size_t)(row0 + v + 8 * hi) * H + m] = acc[v] + badd;
        }
    }
}

// ---------------------------------------------------------------------------
// Kernel 2: out[b][n][:] = relu(zh[b][:] + xh[n][:])  -- pure streaming store.
// One thread per (b,n): 32 contiguous output bytes via 2x b128 NT stores.
// zh is block-uniform (L2 hit), xh coalesced (L2 hit); HBM traffic ~= 64 MiB
// of write-once output, so use non-temporal stores.
// ---------------------------------------------------------------------------
__global__ void __launch_bounds__(256) bcast_relu(
    const float* __restrict__ zh, const float* __restrict__ xh,
    float* __restrict__ out, int N)
{
    const int n = blockIdx.x * 256 + threadIdx.x;
    const int b = blockIdx.y;

    const v4f* __restrict__ zp = (const v4f*)(zh + (size_t)b * H);
    const v4f* __restrict__ xp = (const v4f*)(xh + (size_t)n * H);
    v4f z0 = zp[0], z1 = zp[1];
    v4f x0 = xp[0], x1 = xp[1];

    v4f o0 = z0 + x0;
    v4f o1 = z1 + x1;
    o0.x = fmaxf(o0.x, 0.f); o0.y = fmaxf(o0.y, 0.f);
    o0.z = fmaxf(o0.z, 0.f); o0.w = fmaxf(o0.w, 0.f);
    o1.x = fmaxf(o1.x, 0.f); o1.y = fmaxf(o1.y, 0.f);
    o1.z = fmaxf(o1.z, 0.f); o1.w = fmaxf(o1.w, 0.f);

    v4f* op = (v4f*)(out + ((size_t)b * N + n) * H);
    __builtin_nontemporal_store(o0, op);
    __builtin_nontemporal_store(o1, op + 1);
}

// ---------------------------------------------------------------------------
extern "C" void kernel_launch(void* const* d_in, const int* in_sizes, int n_in,
                              void* d_out, int out_size, void* d_ws, size_t ws_size,
                              hipStream_t stream)
{
    const float* z    = (const float*)d_in[0];   // (B, 64)
    const float* x    = (const float*)d_in[1];   // (N, 64)
    const float* W    = (const float*)d_in[2];   // (128, 8)
    const float* bias = (const float*)d_in[3];   // (8,)
    float* out = (float*)d_out;                  // (B, N, 8)

    const int B = in_sizes[0] / DZ;   // 1024
    const int N = in_sizes[1] / DZ;   // 2048

    float* zh = (float*)d_ws;                    // B*8 floats  (32 KB)
    float* xh = zh + (size_t)B * H;              // N*8 floats  (64 KB)

    const int btiles = B / 16;                   // 64
    const int ntiles = N / 16;                   // 128
    proj_wmma<<<btiles + ntiles, 32, 0, stream>>>(z, x, W, bias, zh, xh, btiles);

    dim3 grid(N / 256, B);                       // (8, 1024)
    bcast_relu<<<grid, 256, 0, stream>>>(zh, xh, out, N);
}